// GlobalLossDminus_28862180229913
// MI455X (gfx1250) — compile-verified
//
#include <hip/hip_runtime.h>
#include <hip/hip_bf16.h>

typedef float v2f __attribute__((ext_vector_type(2)));
typedef float v8f __attribute__((ext_vector_type(8)));

#define NTOT 12288
#define DIM 128
#define BSZ 4096
#define ROWS_PER_BLK 64
#define COLS_PER_TILE 64
#define LDSTRIDE 132            // 128 + 4 pad: conflict-free b64 LDS reads
#define COL_SPLITS 4
#define COLS_PER_SPLIT (NTOT / COL_SPLITS)   // 3072
#define WS_FLOATS (2 * NTOT + 3 * BSZ + 16)

// ---------------- zero workspace ----------------
__global__ __launch_bounds__(256) void zero_kernel(float* __restrict__ p, int n) {
    int i = blockIdx.x * 256 + threadIdx.x;
    if (i < n) p[i] = 0.0f;
}

// ---------------- row L2 norms ----------------
__global__ __launch_bounds__(256) void norms_kernel(const float* __restrict__ X,
                                                    float* __restrict__ norms) {
    int wid  = (blockIdx.x * blockDim.x + threadIdx.x) >> 5;
    int lane = threadIdx.x & 31;
    if (wid >= NTOT) return;
    float4 v = *reinterpret_cast<const float4*>(X + (size_t)wid * DIM + lane * 4);
    float s = v.x * v.x + v.y * v.y + v.z * v.z + v.w * v.w;
    for (int off = 16; off > 0; off >>= 1) s += __shfl_xor(s, off);
    if (lane == 0) norms[wid] = sqrtf(s);
}

// ---------------- Gram tiles + fused exp/mask/row-sum ----------------
// Block = 256 thr (8 waves) owns 64 rows. A fragments live in registers
// (32 x v2f per lane); the single LDS buffer stages A once, then is reused
// for 64-wide B tiles. Each wave computes a 16x32 output region with two
// independent WMMA accumulator chains sharing the register-resident A.
__global__ __launch_bounds__(256) void gram_kernel(const float* __restrict__ X,
                                                   const float* __restrict__ norms,
                                                   float* __restrict__ den) {
    __shared__ float shBuf[COLS_PER_TILE * LDSTRIDE];   // 33792 B, staged A then B
    __shared__ float sNj[COLS_PER_TILE];
    __shared__ float sDen[ROWS_PER_BLK];

    const int tid   = threadIdx.x;
    const int r0    = blockIdx.x * ROWS_PER_BLK;
    const int cbase = blockIdx.y * COLS_PER_SPLIT;

    // stage the 64 anchor rows (64 x 128 f32) into padded LDS
    for (int f = tid; f < ROWS_PER_BLK * (DIM / 4); f += 256) {
        int row = f >> 5;           // 32 float4 per row
        int c4  = f & 31;
        float4 v = *reinterpret_cast<const float4*>(X + (size_t)(r0 + row) * DIM + c4 * 4);
        float* dst = &shBuf[row * LDSTRIDE + c4 * 4];
        dst[0] = v.x; dst[1] = v.y; dst[2] = v.z; dst[3] = v.w;
    }
    if (tid < ROWS_PER_BLK) sDen[tid] = 0.0f;

    const int lane = tid & 31;
    const int w    = tid >> 5;
    const int lr0  = (w & 3) * 16;       // row subtile within 64-row block
    const int lc0  = (w >> 2) * 32;      // col pair base within 64-col tile
    const int l16  = lane & 15;
    const int hi2  = (lane & 16) >> 3;   // +2 dwords for lanes 16..31 (K=2,3)
    const int hi8  = (lane & 16) >> 1;   // +8 row offset for lanes 16..31

    const float nrow_vec = norms[r0 + lr0 + l16];

    __syncthreads();
    // pull this wave's A fragment into registers: reused for the whole column loop
    v2f areg[DIM / 4];
    #pragma unroll
    for (int kk = 0; kk < DIM / 4; ++kk)
        areg[kk] = *reinterpret_cast<const v2f*>(
            &shBuf[(lr0 + l16) * LDSTRIDE + kk * 4 + hi2]);

    v8f rowsum = {0.f, 0.f, 0.f, 0.f, 0.f, 0.f, 0.f, 0.f};

    for (int ct = 0; ct < COLS_PER_SPLIT / COLS_PER_TILE; ++ct) {
        const int c0 = cbase + ct * COLS_PER_TILE;
        __syncthreads();   // A-reg fill done (ct=0) / prior B readers done
        for (int f = tid; f < COLS_PER_TILE * (DIM / 4); f += 256) {
            int row = f >> 5;
            int c4  = f & 31;
            float4 v = *reinterpret_cast<const float4*>(X + (size_t)(c0 + row) * DIM + c4 * 4);
            float* dst = &shBuf[row * LDSTRIDE + c4 * 4];
            dst[0] = v.x; dst[1] = v.y; dst[2] = v.z; dst[3] = v.w;
        }
        if (tid < COLS_PER_TILE) sNj[tid] = norms[c0 + tid];
        __syncthreads();

        v8f acc0 = {0.f, 0.f, 0.f, 0.f, 0.f, 0.f, 0.f, 0.f};
        v8f acc1 = {0.f, 0.f, 0.f, 0.f, 0.f, 0.f, 0.f, 0.f};
        const float* pb0 = &shBuf[(lc0 + l16) * LDSTRIDE + hi2];
        const float* pb1 = pb0 + 16 * LDSTRIDE;
        #pragma unroll
        for (int kk = 0; kk < DIM / 4; ++kk) {
            v2f b0 = *reinterpret_cast<const v2f*>(pb0 + kk * 4);
            v2f b1 = *reinterpret_cast<const v2f*>(pb1 + kk * 4);
            acc0 = __builtin_amdgcn_wmma_f32_16x16x4_f32(
                false, areg[kk], false, b0, (short)0, acc0, false, false);
            acc1 = __builtin_amdgcn_wmma_f32_16x16x4_f32(
                false, areg[kk], false, b1, (short)0, acc1, false, false);
        }

        // epilogue: S = dot * 10 / max(ni*nj, eps); e = exp(S); mask (i%8 != j%8)
        const float nj0 = sNj[lc0 + l16];
        const float nj1 = sNj[lc0 + 16 + l16];
        #pragma unroll
        for (int r = 0; r < 8; ++r) {
            float ni = __shfl(nrow_vec, r + hi8);   // row r (lo half) / r+8 (hi half)
            float s0 = acc0[r] * (10.0f / fmaxf(ni * nj0, 1e-6f));
            float s1 = acc1[r] * (10.0f / fmaxf(ni * nj1, 1e-6f));
            float e  = __expf(s0) + __expf(s1);
            // 16-aligned tiles, M=8 | 16  =>  static mask: j%8 == lane&7, i%8 == r
            if ((lane & 7) != r) rowsum[r] += e;
        }
    }

    // reduce each rowsum[r] across the 16-lane halves, combine waves in LDS
    #pragma unroll
    for (int r = 0; r < 8; ++r) {
        float v = rowsum[r];
        v += __shfl_xor(v, 1);
        v += __shfl_xor(v, 2);
        v += __shfl_xor(v, 4);
        v += __shfl_xor(v, 8);
        if (l16 == 0) atomicAdd(&sDen[lr0 + r + hi8], v);   // ds_add_f32
    }
    __syncthreads();
    if (tid < ROWS_PER_BLK) atomicAdd(&den[r0 + tid], sDen[tid]);  // combine column splits
}

// ---------------- the 3*BS pair similarities ----------------
__global__ __launch_bounds__(256) void pairs_kernel(const float* __restrict__ X,
                                                    const float* __restrict__ norms,
                                                    float* __restrict__ pairS) {
    int q    = (blockIdx.x * blockDim.x + threadIdx.x) >> 5;
    int lane = threadIdx.x & 31;
    if (q >= 3 * BSZ) return;
    int t = q / BSZ, p = q % BSZ;
    int a, b;
    if (t == 0)      { a = p;       b = BSZ + p;     }
    else if (t == 1) { a = p;       b = 2 * BSZ + p; }
    else             { a = BSZ + p; b = 2 * BSZ + p; }
    float4 va = *reinterpret_cast<const float4*>(X + (size_t)a * DIM + lane * 4);
    float4 vb = *reinterpret_cast<const float4*>(X + (size_t)b * DIM + lane * 4);
    float s = va.x * vb.x + va.y * vb.y + va.z * vb.z + va.w * vb.w;
    for (int off = 16; off > 0; off >>= 1) s += __shfl_xor(s, off);
    if (lane == 0) {
        float dnm = fmaxf(norms[a] * norms[b], 1e-6f);
        pairS[q] = s * (10.0f / dnm);
    }
}

// ---------------- per-p loss + block partials ----------------
__global__ __launch_bounds__(256) void loss_kernel(const float* __restrict__ den,
                                                   const float* __restrict__ pairS,
                                                   float* __restrict__ partials) {
    __shared__ float red[256];
    int p = blockIdx.x * 256 + threadIdx.x;
    float d1 = den[p], d2 = den[BSZ + p], d3 = den[2 * BSZ + p];
    float s12 = pairS[p], s13 = pairS[BSZ + p], s23 = pairS[2 * BSZ + p];
    float e12 = __expf(s12), e13 = __expf(s13), e23 = __expf(s23);
    // -log(e/(e+den)) = log(e+den) - s
    float loss = (logf(e12 + d1) - s12) + (logf(e12 + d2) - s12)
               + (logf(e13 + d1) - s13) + (logf(e13 + d3) - s13)
               + (logf(e23 + d2) - s23) + (logf(e23 + d3) - s23);
    red[threadIdx.x] = loss;
    __syncthreads();
    for (int off = 128; off > 0; off >>= 1) {
        if (threadIdx.x < off) red[threadIdx.x] += red[threadIdx.x + off];
        __syncthreads();
    }
    if (threadIdx.x == 0) partials[blockIdx.x] = red[0];
}

__global__ void final_kernel(const float* __restrict__ partials, float* __restrict__ out) {
    float s = 0.0f;
    for (int i = 0; i < BSZ / 256; ++i) s += partials[i];
    out[0] = s / (2.0f * BSZ);   // N_TRANSFORMS * BS = 8192
}

extern "C" void kernel_launch(void* const* d_in, const int* in_sizes, int n_in,
                              void* d_out, int out_size, void* d_ws, size_t ws_size,
                              hipStream_t stream) {
    const float* X = (const float*)d_in[0];
    float* ws       = (float*)d_ws;
    float* norms    = ws;                       // NTOT
    float* den      = ws + NTOT;                // NTOT (accumulated -> must be zeroed)
    float* pairS    = ws + 2 * NTOT;            // 3*BSZ
    float* partials = ws + 2 * NTOT + 3 * BSZ;  // 16
    float* out      = (float*)d_out;

    zero_kernel<<<(WS_FLOATS + 255) / 256, 256, 0, stream>>>(ws, WS_FLOATS);
    norms_kernel<<<NTOT / 8, 256, 0, stream>>>(X, norms);
    gram_kernel<<<dim3(NTOT / ROWS_PER_BLK, COL_SPLITS), 256, 0, stream>>>(X, norms, den);
    pairs_kernel<<<(3 * BSZ) / 8, 256, 0, stream>>>(X, norms, pairS);
    loss_kernel<<<BSZ / 256, 256, 0, stream>>>(den, pairS, partials);
    final_kernel<<<1, 1, 0, stream>>>(partials, out);
}